// NormalVariationBoundaryLoss_19189913879105
// MI455X (gfx1250) — compile-verified
//
#include <hip/hip_runtime.h>
#include <hip/hip_bf16.h>
#include <math.h>

// ---------------------------------------------------------------------------
// NormalVariationBoundaryLoss for MI455X (gfx1250, wave32).
// Passes:
//   0) init hash table + scalars
//   1) hash-insert (atomicCAS on 64b keys) + masked atomic accumulate of
//      normal sums / counts per voxel slot
//   2) per-point boundary strength diff = 1 - cos(n, voxel_mean_n),
//      wave-reduced atomic min/max (uint bit ordering, diff >= 0)
//   3) CE via WMMA f32 16x16x4 row-sum of exp + weighted accumulation
//   4) finalize scalar loss
// ---------------------------------------------------------------------------

typedef __attribute__((ext_vector_type(2))) float v2f;
typedef __attribute__((ext_vector_type(8))) float v8f;

#define EMPTY_KEY 0xFFFFFFFFFFFFFFFFull
#define IGNORE_INDEX (-1)
#define EPS_F 1e-6f
#define BSCALE 3.0f

__global__ void nvbl_init(unsigned long long* __restrict__ keys,
                          float4* __restrict__ acc,
                          unsigned* __restrict__ scal,
                          int cap) {
  int i = blockIdx.x * blockDim.x + threadIdx.x;
  if (i < cap) {
    keys[i] = EMPTY_KEY;
    acc[i] = make_float4(0.f, 0.f, 0.f, 0.f);
  }
  if (i == 0) {
    scal[0] = 0x7F800000u;            // running min bits  (+inf)
    scal[1] = 0x00000000u;            // running max bits  (0.0f; diff >= 0)
    ((float*)scal)[2] = 0.f;          // sum(ce*w*vm)
    ((float*)scal)[3] = 0.f;          // sum(vm)
  }
}

__global__ void nvbl_hash_accum(const int* __restrict__ gc,
                                const int* __restrict__ batch,
                                const int* __restrict__ target,
                                const float* __restrict__ normal,
                                unsigned long long* __restrict__ keys,
                                float4* __restrict__ acc,
                                int* __restrict__ slot,
                                int n, int cap) {
  int i = blockIdx.x * blockDim.x + threadIdx.x;
  if (i >= n) return;
  const long long B1 = 1000003LL, B2 = 10007LL;
  long long x = gc[3 * i + 0], y = gc[3 * i + 1], z = gc[3 * i + 2];
  long long b = batch[i];
  long long vid = x * B1 * B1 + y * B1 + z + b * (B1 * B1 * B1 + B2);
  unsigned long long key = (unsigned long long)vid;
  unsigned h = (unsigned)((key * 0x9E3779B97F4A7C15ull) >> 32) & (unsigned)(cap - 1);
  int s = 0;
  for (int probe = 0; probe < cap; ++probe) {
    unsigned long long prev = atomicCAS(&keys[h], EMPTY_KEY, key);
    if (prev == EMPTY_KEY || prev == key) { s = (int)h; break; }
    h = (h + 1) & (unsigned)(cap - 1);
  }
  slot[i] = s;
  if (target[i] != IGNORE_INDEX) {
    float nx = normal[3 * i + 0], ny = normal[3 * i + 1], nz = normal[3 * i + 2];
    atomicAdd(&acc[s].x, nx);
    atomicAdd(&acc[s].y, ny);
    atomicAdd(&acc[s].z, nz);
    atomicAdd(&acc[s].w, 1.0f);
  }
}

__global__ void nvbl_diff(const int* __restrict__ target,
                          const float* __restrict__ normal,
                          const float4* __restrict__ acc,
                          const int* __restrict__ slot,
                          float* __restrict__ diff,
                          unsigned* __restrict__ scal,
                          int n) {
  int i = blockIdx.x * blockDim.x + threadIdx.x;
  int lane = threadIdx.x & 31;
  bool inb = i < n;
  float dmin = INFINITY, dmax = 0.0f;
  if (inb) {
    float4 a = acc[slot[i]];
    float invc = 1.0f / fmaxf(a.w, 1.0f);
    float mx = a.x * invc, my = a.y * invc, mz = a.z * invc;
    float mnorm = sqrtf(mx * mx + my * my + mz * mz);
    float im = 1.0f / fmaxf(mnorm, EPS_F);
    mx *= im; my *= im; mz *= im;
    float nx = normal[3 * i + 0], ny = normal[3 * i + 1], nz = normal[3 * i + 2];
    float nn = sqrtf(nx * nx + ny * ny + nz * nz);
    float in_ = 1.0f / fmaxf(nn, EPS_F);
    float cosv = (nx * mx + ny * my + nz * mz) * in_;
    cosv = fminf(1.0f, fmaxf(-1.0f, cosv));
    float d = 1.0f - cosv;
    diff[i] = d;
    if (target[i] != IGNORE_INDEX) { dmin = d; dmax = d; }
  }
  // wave32 min/max reduce (diff >= 0, so uint ordering == float ordering)
  for (int o = 16; o; o >>= 1) {
    dmin = fminf(dmin, __shfl_xor(dmin, o, 32));
    dmax = fmaxf(dmax, __shfl_xor(dmax, o, 32));
  }
  if (lane == 0) {
    atomicMin(&scal[0], __float_as_uint(dmin));
    atomicMax(&scal[1], __float_as_uint(dmax));
  }
}

// One wave per 16 rows; 8 chained V_WMMA_F32_16X16X4_F32 compute exact-f32
// row sums of exp(x - max) (B = ones, every D column replicates the row sum).
__global__ void nvbl_ce_wmma(const float* __restrict__ pred,
                             const int* __restrict__ target,
                             const float* __restrict__ diff,
                             const unsigned* __restrict__ scal,
                             float* __restrict__ sums,
                             int n) {
  const int tid = blockIdx.x * blockDim.x + threadIdx.x;
  const int lane = threadIdx.x & 31;
  const int half = lane >> 4;       // 0: K=0,1  1: K=2,3 of each 4-class chunk
  const int r = lane & 15;          // row within the 16-row tile
  const int waveId = tid >> 5;
  const int nWaves = (gridDim.x * blockDim.x) >> 5;
  const int ngroups = (n + 15) >> 4;

  const float dmin = __uint_as_float(scal[0]);
  const float dmax = __uint_as_float(scal[1]);
  const float invRange = 1.0f / (dmax - dmin + EPS_F);

  v2f ones; ones[0] = 1.0f; ones[1] = 1.0f;

  for (int g = waveId; g < ngroups; g += nWaves) {
    int row_raw = (g << 4) + r;
    bool rowOK = row_raw < n;
    int row = rowOK ? row_raw : (n - 1);
    const float* prow = pred + (size_t)row * 32 + half * 2;

    // Prefetch this wave's next tile into L2 (192 MB on MI455X).
    int gnext = g + nWaves;
    if (gnext < ngroups)
      __builtin_prefetch(pred + ((size_t)(gnext) << 4) * 32 + (size_t)r * 32, 0, 1);

    float2 v[8];
    float m = -INFINITY;
#pragma unroll
    for (int c = 0; c < 8; ++c) {
      v[c] = *(const float2*)(prow + c * 4);
      m = fmaxf(m, fmaxf(v[c].x, v[c].y));
    }
    m = fmaxf(m, __shfl_xor(m, 16, 32));   // full row max across the two halves

    v8f d = {};
#pragma unroll
    for (int c = 0; c < 8; ++c) {
      v2f a;
      a[0] = __expf(v[c].x - m);
      a[1] = __expf(v[c].y - m);
      d = __builtin_amdgcn_wmma_f32_16x16x4_f32(
          /*neg_a=*/false, a, /*neg_b=*/false, ones,
          /*c_mod=*/(short)0, d, /*reuse_a=*/false, /*reuse_b=*/false);
    }

    // Extract row sum for row r: rows 0-7 live in lanes 0-15 (VGPR rl),
    // rows 8-15 in lanes 16-31 (VGPR rl).  cndmask tree + one shfl.
    int rl = r & 7;
    float t01 = (rl & 1) ? d[1] : d[0];
    float t23 = (rl & 1) ? d[3] : d[2];
    float t45 = (rl & 1) ? d[5] : d[4];
    float t67 = (rl & 1) ? d[7] : d[6];
    float t0123 = (rl & 2) ? t23 : t01;
    float t4567 = (rl & 2) ? t67 : t45;
    float t = (rl & 4) ? t4567 : t0123;
    int srcLane = (r < 8) ? r : (r + 8);
    float S = __shfl(t, srcLane, 32);

    int tgt = target[row];
    bool valid = rowOK && (tgt != IGNORE_INDEX);
    int tsafe = (tgt >= 0 && tgt < 32) ? tgt : 0;
    float xt = pred[(size_t)row * 32 + tsafe];        // cached (row just read)
    float ce = m + logf(S) - xt;                      // -log_softmax[target]

    float dd = diff[row];
    float dn = fminf(1.0f, fmaxf(0.0f, (dd - dmin) * invRange));
    float w = 1.0f + BSCALE * dn;

    float contrib = (valid && half == 0) ? ce * w : 0.0f;
    float cm = (valid && half == 0) ? 1.0f : 0.0f;
    for (int o = 16; o; o >>= 1) {
      contrib += __shfl_xor(contrib, o, 32);
      cm += __shfl_xor(cm, o, 32);
    }
    if (lane == 0) {
      atomicAdd(&sums[0], contrib);
      atomicAdd(&sums[1], cm);
    }
  }
}

__global__ void nvbl_finalize(const float* __restrict__ sums,
                              float* __restrict__ out) {
  if (threadIdx.x == 0 && blockIdx.x == 0)
    out[0] = sums[0] / fmaxf(sums[1], 1.0f);   // * LOSS_WEIGHT (1.0)
}

extern "C" void kernel_launch(void* const* d_in, const int* in_sizes, int n_in,
                              void* d_out, int out_size, void* d_ws, size_t ws_size,
                              hipStream_t stream) {
  const float* pred   = (const float*)d_in[0];   // (N, 32) f32
  const int*   target = (const int*)d_in[1];     // (N,)
  const int*   gc     = (const int*)d_in[2];     // (N, 3)
  const float* normal = (const float*)d_in[3];   // (N, 3) f32
  const int*   batch  = (const int*)d_in[4];     // (N,)
  const int n = in_sizes[1];

  // ---- workspace layout (hash table sized to fit ws) ----
  size_t cap = (size_t)1 << 22;                  // up to 4M slots (load ~0.47)
  while (cap > ((size_t)1 << 20) &&
         cap * 8 + cap * 16 + (size_t)n * 8 + 256 > ws_size)
    cap >>= 1;
  char* base = (char*)d_ws;
  size_t off = 0;
  unsigned long long* keys = (unsigned long long*)(base + off); off += cap * 8;
  float4* acc = (float4*)(base + off);                          off += cap * 16;
  int* slot = (int*)(base + off);    off += ((size_t)n * 4 + 15) & ~(size_t)15;
  float* diff = (float*)(base + off); off += ((size_t)n * 4 + 15) & ~(size_t)15;
  unsigned* scal = (unsigned*)(base + off);      // 4 dwords: min,max,sumL,sumM
  float* sums = (float*)scal + 2;

  const int icap = (int)cap;
  dim3 blk(256);

  nvbl_init<<<dim3((icap + 255) / 256), blk, 0, stream>>>(keys, acc, scal, icap);
  nvbl_hash_accum<<<dim3((n + 255) / 256), blk, 0, stream>>>(
      gc, batch, target, normal, keys, acc, slot, n, icap);
  nvbl_diff<<<dim3((n + 255) / 256), blk, 0, stream>>>(
      target, normal, acc, slot, diff, scal, n);

  int ngroups = (n + 15) >> 4;                   // 16 rows per wave
  int blocks = (ngroups + 7) / 8;                // 8 waves / 256-thread block
  if (blocks < 1) blocks = 1;
  nvbl_ce_wmma<<<dim3(blocks), blk, 0, stream>>>(
      pred, target, diff, scal, sums, n);

  nvbl_finalize<<<dim3(1), dim3(32), 0, stream>>>(sums, (float*)d_out);
}